// Net_24026047054166
// MI455X (gfx1250) — compile-verified
//
#include <hip/hip_runtime.h>
#include <hip/hip_bf16.h>

typedef float v2f __attribute__((ext_vector_type(2)));
typedef float v8f __attribute__((ext_vector_type(8)));

#define TILE 32
#define XS_W 37   // 36-wide input tile, +1 pad
#define ZS_W 38   // 34-wide intermediate, +4 (guaranteed zeros for K=18,19)

// Compile-time image geometry (reference fixes 16x1x2048x2048).
#define IMG_H 2048
#define IMG_W 2048

// Low 32 bits of a generic pointer to __shared__ = LDS byte offset (ISA §10.2).
#define LDS_OFF(p) ((unsigned)(unsigned long long)(const volatile void*)(p))

__global__ __launch_bounds__(128) void conv3x3x2_fused(
    const float* __restrict__ xin, const float* __restrict__ w1,
    const float* __restrict__ w2, float* __restrict__ out)
{
    __shared__ float xs[36 * XS_W];
    __shared__ float zs[34 * ZS_W];

    const int tid  = threadIdx.x;
    const int bx   = blockIdx.x, by = blockIdx.y, bimg = blockIdx.z;
    const long long base = (long long)bimg * (IMG_H * IMG_W);
    const int x0 = bx * TILE, y0 = by * TILE;

    // conv1 weights, uniform -> scalar regs
    const float k0 = w1[0], k1 = w1[1], k2 = w1[2],
                k3 = w1[3], k4 = w1[4], k5 = w1[5],
                k6 = w1[6], k7 = w1[7], k8 = w1[8];

    // Input tile spans rows [y0-2, y0+33], cols [x0-2, x0+33].
    const bool interior = (x0 >= 2) && (y0 >= 2) &&
                          (x0 + 34 <= IMG_W) && (y0 + 34 <= IMG_H);

    // ---- Stage 0: stage 36x36 input tile into LDS ----
    if (interior) {
        // Async DMA global -> LDS (ASYNCcnt path), branch-free.
        const float* src = xin + base + (long long)(y0 - 2) * IMG_W + (x0 - 2);
        for (int idx = tid; idx < 36 * 36; idx += 128) {
            const int r = idx / 36, c = idx - r * 36;
            const float* g = src + r * IMG_W + c;
            unsigned lds = LDS_OFF(&xs[r * XS_W + c]);
            asm volatile("global_load_async_to_lds_b32 %0, %1, off"
                         :: "v"(lds), "v"(g) : "memory");
        }
    } else {
        for (int idx = tid; idx < 36 * 36; idx += 128) {
            const int r = idx / 36, c = idx - r * 36;
            const int iy = y0 + r - 2, ix = x0 + c - 2;
            float v = 0.f;
            if (iy >= 0 && iy < IMG_H && ix >= 0 && ix < IMG_W)
                v = xin[base + (long long)iy * IMG_W + ix];
            xs[r * XS_W + c] = v;
        }
    }
    // zero the 4 guard columns of zs (cols 34..37) so WMMA K=18,19 reads exact zeros
    for (int idx = tid; idx < 34 * 4; idx += 128) {
        const int r = idx >> 2, c = 34 + (idx & 3);
        zs[r * ZS_W + c] = 0.f;
    }
    if (interior)
        asm volatile("s_wait_asynccnt 0" ::: "memory");
    __syncthreads();

    // ---- Stage 1: conv1 -> 34x34 intermediate in LDS (zero outside image:
    //      reproduces conv2's zero-padded-intermediate semantics exactly) ----
    if (interior) {
        for (int idx = tid; idx < 34 * 34; idx += 128) {
            const int zr = idx / 34, zc = idx - zr * 34;
            const float* p = &xs[zr * XS_W + zc];
            zs[zr * ZS_W + zc] =
                  k0 * p[0]        + k1 * p[1]          + k2 * p[2]
                + k3 * p[XS_W]     + k4 * p[XS_W + 1]   + k5 * p[XS_W + 2]
                + k6 * p[2*XS_W]   + k7 * p[2*XS_W + 1] + k8 * p[2*XS_W + 2];
        }
    } else {
        for (int idx = tid; idx < 34 * 34; idx += 128) {
            const int zr = idx / 34, zc = idx - zr * 34;
            const int iy = y0 + zr - 1, ix = x0 + zc - 1;
            float acc = 0.f;
            if (iy >= 0 && iy < IMG_H && ix >= 0 && ix < IMG_W) {
                const float* p = &xs[zr * XS_W + zc];
                acc = k0 * p[0]        + k1 * p[1]          + k2 * p[2]
                    + k3 * p[XS_W]     + k4 * p[XS_W + 1]   + k5 * p[XS_W + 2]
                    + k6 * p[2*XS_W]   + k7 * p[2*XS_W + 1] + k8 * p[2*XS_W + 2];
            }
            zs[zr * ZS_W + zc] = acc;
        }
    }
    __syncthreads();

    // ---- Stage 2: conv2 as banded matmul via V_WMMA_F32_16X16X4_F32 ----
    //   Y(16x16) = sum_{di=0..2} A_di(16x20) * B_di(20x16)
    //   A_di[i,c'] = zs[i0+i+di][j0+c'],  B_di[c',j] = (0<=c'-j<=2 && c'<18) ? w2[di][c'-j] : 0
    const int wave = tid >> 5, lane = tid & 31;
    const int i0 = (wave >> 1) * 16, j0 = (wave & 1) * 16;
    const int m  = lane & 15;         // A row / B,D column
    const int hi = lane >> 4;         // half-lane group: K offset +2 (A,B), M offset +8 (D)

    // Banded B fragments: depend only on w2 and lane -> 15 v2f registers
    v2f bfrag[3][5];
#pragma unroll
    for (int di = 0; di < 3; ++di) {
#pragma unroll
        for (int kc = 0; kc < 5; ++kc) {
            const int cb = kc * 4 + hi * 2;           // K row of .x ; .y is cb+1
            const int d0 = cb - m, d1 = cb + 1 - m;
            bfrag[di][kc].x = (cb     < 18 && d0 >= 0 && d0 <= 2) ? w2[di * 3 + d0] : 0.f;
            bfrag[di][kc].y = (cb + 1 < 18 && d1 >= 0 && d1 <= 2) ? w2[di * 3 + d1] : 0.f;
        }
    }

    v8f acc = {};
#pragma unroll
    for (int di = 0; di < 3; ++di) {
#pragma unroll
        for (int kc = 0; kc < 5; ++kc) {
            const int zrow = i0 + m + di;
            const int zcol = j0 + kc * 4 + hi * 2;    // even -> 8B-aligned ds_load_b64
            v2f a;
            a.x = zs[zrow * ZS_W + zcol];
            a.y = zs[zrow * ZS_W + zcol + 1];
            acc = __builtin_amdgcn_wmma_f32_16x16x4_f32(
                false, a, false, bfrag[di][kc], (short)0, acc, false, false);
        }
    }

    // D layout: VGPR r -> row i0 + r + hi*8, col j0 + m.
    // IMG_W constant => po[r*IMG_W] folds to one base address + 8 stores with
    // immediate offsets (r*8192 B fits the signed 24-bit IOFFSET).
    float* po = out + base + (long long)(y0 + i0 + hi * 8) * IMG_W + (x0 + j0 + m);
#pragma unroll
    for (int r = 0; r < 8; ++r)
        po[r * IMG_W] = acc[r];
}

extern "C" void kernel_launch(void* const* d_in, const int* in_sizes, int n_in,
                              void* d_out, int out_size, void* d_ws, size_t ws_size,
                              hipStream_t stream) {
    const float* x  = (const float*)d_in[0];
    const float* w1 = (const float*)d_in[1];
    const float* w2 = (const float*)d_in[2];
    float* out = (float*)d_out;

    const int B = in_sizes[0] / (IMG_H * IMG_W);

    dim3 grid(IMG_W / TILE, IMG_H / TILE, B);
    dim3 block(128);
    conv3x3x2_fused<<<grid, block, 0, stream>>>(x, w1, w2, out);
}